// MovingAverage_75033078661801
// MI455X (gfx1250) — compile-verified
//
#include <hip/hip_runtime.h>

typedef float v2f __attribute__((ext_vector_type(2)));
typedef float v8f __attribute__((ext_vector_type(8)));

#define T_TOTAL 1048576
#define NF 64          // features
#define NK 16          // selected columns
#define WIN 128        // window
#define TB  256        // time rows per workgroup
#define NCH 16         // main chunks per block (TB/16)
#define HCH 8          // halo chunks (WIN/16)
#define TCH 24         // total chunks
#define LSTR 17        // padded LDS row stride (floats)

__global__ __launch_bounds__(256)
void MovingAverage_75033078661801_kernel(const float* __restrict__ inp,
                                         const int* __restrict__ feats,
                                         float* __restrict__ out)
{
    __shared__ float gat[TCH * 16 * LSTR];   // gathered selected cols [chunk][r][k]
    __shared__ float pre[TCH * 16 * LSTR];   // exclusive intra-chunk prefix
    __shared__ float tot[TCH * LSTR];        // chunk totals [chunk][k]
    __shared__ float cs8[NCH * LSTR];        // sum of 8 preceding chunk totals
    __shared__ int   fcols[NK];
    __shared__ int   colmap[NF];

    const int  tid = threadIdx.x;
    const long t0  = (long)blockIdx.x * TB;

    // ---- column map ----
    if (tid < NF) colmap[tid] = -1;
    if (tid < NK) fcols[tid] = feats[tid];
    __syncthreads();
    if (tid < NK) colmap[fcols[tid]] = tid;

    // ---- Phase A: gather selected columns into LDS ----
    {   // main rows: one per thread
        const long   t  = t0 + tid;
        const int    ch = HCH + (tid >> 4), r = tid & 15;
        const float* rp = inp + t * NF;
        float*       g  = &gat[(ch * 16 + r) * LSTR];
        #pragma unroll
        for (int k = 0; k < NK; ++k) g[k] = rp[fcols[k]];
    }
    if (tid < WIN) { // halo rows t0-128 .. t0-1
        const long th = t0 - WIN + tid;
        const int  ch = tid >> 4, r = tid & 15;
        float*     g  = &gat[(ch * 16 + r) * LSTR];
        if (th >= 0) {
            const float* rp = inp + th * NF;
            #pragma unroll
            for (int k = 0; k < NK; ++k) g[k] = rp[fcols[k]];
        } else {
            #pragma unroll
            for (int k = 0; k < NK; ++k) g[k] = 0.0f;
        }
    }
    __syncthreads();

    // ---- Phase B: exclusive intra-chunk prefix via fp32 WMMA ----
    // P(16x16) = strict_lower_tri(16x16) * X(16 time x 16 feat), K split 4x4.
    {
        const int wave = tid >> 5;
        const int lane = tid & 31;
        const int half = lane >> 4;     // K/M half selector per ISA layout
        const int n    = lane & 15;     // column (B/D) or row m (A)
        #pragma unroll
        for (int i = 0; i < 3; ++i) {
            const int c = wave + i * 8;          // chunk index (24 chunks / 8 waves)
            v8f acc = {};
            #pragma unroll
            for (int j = 0; j < 4; ++j) {
                const int s0 = 4 * j + 2 * half; // K rows handled by this lane-half
                v2f a, b;
                a.x = (s0     < n) ? 1.0f : 0.0f;  // L_strict[m][s0],   m == n (lane)
                a.y = (s0 + 1 < n) ? 1.0f : 0.0f;  // L_strict[m][s0+1]
                b.x = gat[(c * 16 + s0    ) * LSTR + n];  // X[s0][n]
                b.y = gat[(c * 16 + s0 + 1) * LSTR + n];  // X[s0+1][n]
                acc = __builtin_amdgcn_wmma_f32_16x16x4_f32(
                          false, a, false, b, (short)0, acc, false, false);
            }
            // D layout: VGPR r -> row (r + 8*half), lane -> col n
            #pragma unroll
            for (int r = 0; r < 8; ++r)
                pre[(c * 16 + r + 8 * half) * LSTR + n] = acc[r];
            if (half) // lane holds P[15][n]; chunk total = P_excl[15] + X[15]
                tot[c * LSTR + n] = acc[7] + gat[(c * 16 + 15) * LSTR + n];
        }
    }
    __syncthreads();

    // ---- Phase C: sliding sum of 8 preceding chunk totals ----
    {
        const int oc = tid >> 4, k = tid & 15;
        float s = 0.0f;
        #pragma unroll
        for (int j = 0; j < 8; ++j) s += tot[(oc + j) * LSTR + k];
        cs8[oc * LSTR + k] = s;      // chunks (qb-8 .. qb-1) for qb = oc+8
    }
    __syncthreads();

    // ---- Phase D: stream full rows, substituting MA values ----
    {
        const long   t    = t0 + tid;
        const int    oc   = tid >> 4, r = tid & 15;
        const int    qb   = HCH + oc;
        const bool   doMA = (t >= WIN);
        const float  inv  = 1.0f / (float)WIN;
        const float4* rin  = (const float4*)(inp + t * NF);
        float4*       rout = (float4*)(out + t * NF);
        #pragma unroll
        for (int j = 0; j < NF / 4; ++j) {
            float4 v = rin[j];
            #pragma unroll
            for (int comp = 0; comp < 4; ++comp) {
                const int c = 4 * j + comp;
                const int k = colmap[c];
                if (k >= 0 && doMA) {
                    const float s = cs8[oc * LSTR + k]
                                  + pre[(qb * 16 + r) * LSTR + k]
                                  - pre[(oc * 16 + r) * LSTR + k];
                    ((float*)&v)[comp] = s * inv;
                }
            }
            rout[j] = v;
        }
    }
}

extern "C" void kernel_launch(void* const* d_in, const int* in_sizes, int n_in,
                              void* d_out, int out_size, void* d_ws, size_t ws_size,
                              hipStream_t stream) {
    (void)in_sizes; (void)n_in; (void)out_size; (void)d_ws; (void)ws_size;
    const float* inp   = (const float*)d_in[0];
    const int*   feats = (const int*)d_in[1];   // JAX default x64-off -> int32
    float*       outp  = (float*)d_out;
    dim3 grid(T_TOTAL / TB);
    dim3 block(256);
    MovingAverage_75033078661801_kernel<<<grid, block, 0, stream>>>(inp, feats, outp);
}